// TDLoss_23141283791037
// MI455X (gfx1250) — compile-verified
//
#include <hip/hip_runtime.h>

// GAE backward scan, B rows x S steps, one block per row.
// Memory-bound: 384 MiB total traffic -> ~17 us roofline at 23.3 TB/s.
// CDNA5 path: GLOBAL_LOAD_ASYNC_TO_LDS_B128 staging + s_wait_asynccnt,
// block-parallel affine scan in LDS (mask-reset folded into the operator).

#define S_LEN   2048
#define TPB     256
#define EPT     (S_LEN / TPB)          // 8 elements per thread
#define GAMMA_C 0.999f
#define GL_C    (0.999f * 0.95f)       // gamma * lambda

// Exact vector type the builtin signature uses (per clang diagnostic).
typedef int v4i __attribute__((vector_size(16)));
typedef __attribute__((address_space(1))) v4i* gvp1;   // prints as "__device__"
typedef __attribute__((address_space(3))) v4i* lvp3;   // prints as "__shared__"

// Low 32 bits of a generic pointer to __shared__ data == raw LDS address.
__device__ __forceinline__ unsigned lds_u32(const void* p) {
  return (unsigned)(unsigned long long)p;
}

// Async DMA: 16 bytes per lane, global -> LDS. Tracked by ASYNCcnt.
__device__ __forceinline__ void async_ld16(const void* gsrc, unsigned lds_off) {
#if __has_builtin(__builtin_amdgcn_global_load_async_to_lds_b128)
  __builtin_amdgcn_global_load_async_to_lds_b128(
      (gvp1)(unsigned long long)gsrc, (lvp3)lds_off, 0, 0);
#else
  asm volatile("global_load_async_to_lds_b128 %0, %1, off"
               :: "v"(lds_off), "v"(gsrc)
               : "memory");
#endif
}

__device__ __forceinline__ void wait_async_all() {
#if __has_builtin(__builtin_amdgcn_s_wait_asynccnt)
  __builtin_amdgcn_s_wait_asynccnt(0);
#else
  asm volatile("s_wait_asynccnt 0x0" ::: "memory");
#endif
}

__global__ __launch_bounds__(TPB) void gae_scan_kernel(
    const float* __restrict__ rewards, const float* __restrict__ values,
    const int* __restrict__ dones, const int* __restrict__ mask,
    float* __restrict__ out, int Bn) {
  __shared__ __align__(16) float ldsR[S_LEN];
  __shared__ __align__(16) float ldsV[S_LEN];
  __shared__ __align__(16) int   ldsD[S_LEN];
  __shared__ __align__(16) int   ldsM[S_LEN];
  __shared__ float sA[TPB];
  __shared__ float sB[TPB];

  const int tid = threadIdx.x;
  const int row = blockIdx.x;
  const size_t rbase = (size_t)row * S_LEN;

  const char* gR = (const char*)(rewards + rbase);
  const char* gV = (const char*)(values  + rbase);
  const char* gD = (const char*)(dones   + rbase);
  const char* gM = (const char*)(mask    + rbase);

  const unsigned lR = lds_u32(ldsR);
  const unsigned lV = lds_u32(ldsV);
  const unsigned lD = lds_u32(ldsD);
  const unsigned lM = lds_u32(ldsM);

  // Stage the whole row (4 arrays x 8 KiB) into LDS via async DMA.
  // 2 passes of 256 lanes x 16 B = 8 KiB per array; dense 512B/wave/instr.
#pragma unroll
  for (int c = 0; c < (S_LEN * 4) / (16 * TPB); ++c) {  // 2 iterations
    const int off = (c * TPB + tid) * 16;               // byte offset, 16B aligned
    async_ld16(gR + off, lR + off);
    async_ld16(gV + off, lV + off);
    async_ld16(gD + off, lD + off);
    async_ld16(gM + off, lM + off);
  }
  wait_async_all();     // drain this wave's ASYNCcnt
  __syncthreads();      // all waves' DMA visible to whole block

  // ---- Phase 1: per-thread affine composition over 8 contiguous elements.
  // g[t] = b[t] + a[t]*g[t+1], with (a,b) = mask ? (gl*(1-done), delta) : (0,0)
  const int base = tid * EPT;
  float a[EPT], b[EPT];
  float A = 1.0f, Bc = 0.0f;   // segment: g_left = Bc + A * g_right
#pragma unroll
  for (int k = EPT - 1; k >= 0; --k) {
    const int t = base + k;
    const int m = ldsM[t];
    const float nd = 1.0f - (float)ldsD[t];
    const int mnext = (t + 1 < S_LEN) ? ldsM[t + 1] : 0;
    const float nv = mnext ? ldsV[t + 1] : 0.0f;      // next value, 0 past last valid
    const float delta = ldsR[t] + GAMMA_C * nv * nd - ldsV[t];
    const float ak = m ? (GL_C * nd) : 0.0f;
    const float bk = m ? delta : 0.0f;
    a[k] = ak; b[k] = bk;
    Bc = bk + ak * Bc;         // compose f_k on the left
    A  = ak * A;
  }

  // ---- Phase 2: block-wide backward composition scan (Hillis-Steele).
  sA[tid] = A; sB[tid] = Bc;
  __syncthreads();
#pragma unroll
  for (int d = 1; d < TPB; d <<= 1) {
    float nA = A, nB = Bc;
    if (tid + d < TPB) {                 // compose with segment to the right
      nA = A * sA[tid + d];
      nB = Bc + A * sB[tid + d];
    }
    __syncthreads();
    sA[tid] = nA; sB[tid] = nB;
    A = nA; Bc = nB;
    __syncthreads();
  }
  // Incoming carry for this thread = scan result applied to 0 at row end.
  float g = (tid + 1 < TPB) ? sB[tid + 1] : 0.0f;

  // ---- Phase 3: replay 8 elements with the true carry; produce adv & ret.
  float adv[EPT], ret[EPT];
#pragma unroll
  for (int k = EPT - 1; k >= 0; --k) {
    g = b[k] + a[k] * g;                 // zero automatically outside valid prefix
    adv[k] = g;
    const int t = base + k;
    ret[k] = ldsM[t] ? (g + ldsV[t]) : 0.0f;
  }

  // Coalesced 128-bit stores. d_out = [advantages | returns], each B*S floats.
  float* ap = out + rbase + base;
  float* rp = out + (size_t)Bn * S_LEN + rbase + base;
#pragma unroll
  for (int h = 0; h < EPT / 4; ++h) {
    float4 va; va.x = adv[4*h+0]; va.y = adv[4*h+1]; va.z = adv[4*h+2]; va.w = adv[4*h+3];
    float4 vr; vr.x = ret[4*h+0]; vr.y = ret[4*h+1]; vr.z = ret[4*h+2]; vr.w = ret[4*h+3];
    *reinterpret_cast<float4*>(ap + 4*h) = va;
    *reinterpret_cast<float4*>(rp + 4*h) = vr;
  }
}

extern "C" void kernel_launch(void* const* d_in, const int* in_sizes, int n_in,
                              void* d_out, int out_size, void* d_ws, size_t ws_size,
                              hipStream_t stream) {
  const float* rewards = (const float*)d_in[0];
  const float* values  = (const float*)d_in[1];
  const int*   dones   = (const int*)d_in[2];
  const int*   mask    = (const int*)d_in[3];
  float* out = (float*)d_out;
  const int Bn = in_sizes[0] / S_LEN;   // 8192 rows
  gae_scan_kernel<<<Bn, TPB, 0, stream>>>(rewards, values, dones, mask, out, Bn);
}